// AggNet_59648505806958
// MI455X (gfx1250) — compile-verified
//
#include <hip/hip_runtime.h>
#include <hip/hip_bf16.h>

#define EPS 1e-5f

typedef __attribute__((ext_vector_type(2))) float v2f;
typedef __attribute__((ext_vector_type(8))) float v8f;
typedef __attribute__((ext_vector_type(4))) unsigned int u32x4;
typedef __attribute__((ext_vector_type(8))) int i32x8;
typedef __attribute__((ext_vector_type(4))) int i32x4;

static __device__ __forceinline__ v8f wmma_f32_k4(v2f a, v2f b, v8f c) {
  // D = A(16x4 f32) * B(4x16 f32) + C(16x16 f32)
  return __builtin_amdgcn_wmma_f32_16x16x4_f32(false, a, false, b, (short)0, c,
                                               false, false);
}

static __device__ __forceinline__ v8f vzero8() {
  v8f v;
#pragma unroll
  for (int i = 0; i < 8; ++i) v[i] = 0.0f;
  return v;
}

// ---------------------------------------------------------- TDM weight DMA
// 1-D contiguous copy of `ndw` DWORDs from global to LDS via the Tensor Data
// Mover (D# per CDNA5 ISA ch.8: group0 = control/addr, group1 = dims/strides;
// groups 2-3 zero => <=2D tensor). The LDS byte address is the low 32 bits of
// the shared-aperture flat pointer.
#if __has_builtin(__builtin_amdgcn_tensor_load_to_lds)
#define HAVE_TDM 1
static __device__ __forceinline__ void tdm_load_1d_to_lds(
    const void* gptr, void* lds_ptr, unsigned ndw) {
  unsigned long long ga = (unsigned long long)gptr;
  unsigned lds_off = (unsigned)(unsigned long long)(uintptr_t)lds_ptr;
  u32x4 g0;
  g0[0] = 1u;                                   // count=1 (valid user D#)
  g0[1] = lds_off;                              // lds_addr (bytes)
  g0[2] = (unsigned)(ga & 0xffffffffull);       // global_addr[31:0]
  g0[3] = (unsigned)((ga >> 32) & 0x01ffffffull) | (2u << 30);  // [56:32]+type=2
  i32x8 g1;
  g1[0] = (int)(2u << 16);                      // data_size = 4 bytes
  g1[1] = (int)((ndw & 0xffffu) << 16);         // tensor_dim0[15:0]
  g1[2] = (int)(((ndw >> 16) & 0xffffu) | (1u << 16));  // dim0[31:16], dim1=1
  g1[3] = (int)((ndw & 0xffffu) << 16);         // tile_dim0 = ndw
  g1[4] = 0;                                    // tile_dim1/2 unused
  g1[5] = (int)ndw;                             // tensor_dim0_stride[31:0]
  g1[6] = (int)((ndw & 0xffffu) << 16);         // stride0 hi=0, stride1[15:0]
  g1[7] = 0;
  i32x4 z4 = {0, 0, 0, 0};
#if defined(__clang_major__) && (__clang_major__ >= 23)
  i32x8 z8 = {0, 0, 0, 0, 0, 0, 0, 0};
  __builtin_amdgcn_tensor_load_to_lds(g0, g1, z4, z4, z8, 0);
#else
  __builtin_amdgcn_tensor_load_to_lds(g0, g1, z4, z4, 0);
#endif
}
#else
#define HAVE_TDM 0
#endif

// ---------------------------------------------------------------- utilities
__global__ void fillzero_kernel(float* __restrict__ p, size_t n) {
  for (size_t i = (size_t)blockIdx.x * blockDim.x + threadIdx.x; i < n;
       i += (size_t)gridDim.x * blockDim.x)
    p[i] = 0.0f;
}

// ------------------------------------------------------------- degree / wn
__global__ void deg_kernel(const int* __restrict__ row,
                           const float* __restrict__ w,
                           float* __restrict__ deg, int E) {
  for (size_t e = (size_t)blockIdx.x * blockDim.x + threadIdx.x; e < (size_t)E;
       e += (size_t)gridDim.x * blockDim.x)
    atomicAdd(&deg[row[e]], w[e]);
}

__global__ void wn_kernel(const int* __restrict__ row,
                          const int* __restrict__ col,
                          const float* __restrict__ w,
                          const float* __restrict__ deg,
                          float* __restrict__ wn, int E) {
  for (size_t e = (size_t)blockIdx.x * blockDim.x + threadIdx.x; e < (size_t)E;
       e += (size_t)gridDim.x * blockDim.x) {
    float dr = deg[row[e]];
    float dc = deg[col[e]];
    float ir = dr > 0.0f ? rsqrtf(dr) : 0.0f;
    float ic = dc > 0.0f ? rsqrtf(dc) : 0.0f;
    wn[e] = -ir * w[e] * ic;
  }
}

// ----------------------------------------------------------- instance norm
template <int C>
__global__ void inorm_stats_kernel(const float* __restrict__ x,
                                   float* __restrict__ stats, int N) {
  __shared__ float ss[2 * C];
  for (int i = threadIdx.x; i < 2 * C; i += blockDim.x) ss[i] = 0.0f;
  __syncthreads();
  size_t total = (size_t)N * C;
  for (size_t idx = (size_t)blockIdx.x * blockDim.x + threadIdx.x; idx < total;
       idx += (size_t)gridDim.x * blockDim.x) {
    int c = (int)(idx % C);
    float v = x[idx];
    atomicAdd(&ss[c], v);
    atomicAdd(&ss[C + c], v * v);
  }
  __syncthreads();
  for (int i = threadIdx.x; i < 2 * C; i += blockDim.x)
    atomicAdd(&stats[i], ss[i]);
}

template <int C>
__global__ void inorm_apply_kernel(const float* __restrict__ x,
                                   const float* __restrict__ stats,
                                   float* __restrict__ xn, int N) {
  float invN = 1.0f / (float)N;
  size_t total = (size_t)N * C;
  for (size_t idx = (size_t)blockIdx.x * blockDim.x + threadIdx.x; idx < total;
       idx += (size_t)gridDim.x * blockDim.x) {
    int c = (int)(idx % C);
    float mean = stats[c] * invN;
    float var = stats[C + c] * invN - mean * mean;
    xn[idx] = (x[idx] - mean) * rsqrtf(var + EPS);
  }
}

// --------------------------------------------------------------------- spmm
// out[col[e], c] += wn[e] * x[row[e], c]
template <int C>
__global__ void spmm_kernel(const int* __restrict__ row,
                            const int* __restrict__ col,
                            const float* __restrict__ wn,
                            const float* __restrict__ x,
                            float* __restrict__ out, int E) {
  size_t total = (size_t)E * C;
  for (size_t idx = (size_t)blockIdx.x * blockDim.x + threadIdx.x; idx < total;
       idx += (size_t)gridDim.x * blockDim.x) {
    size_t e = idx / C;
    int c = (int)(idx % C);
    float v = wn[e] * x[(size_t)row[e] * C + c];
    atomicAdd(&out[(size_t)col[e] * C + c], v);
  }
}

// ----------------------------------------------- cheb combine (scalar path)
// out = relu(Tx0@W0 + Tx1@W1 + (2*T2 - Tx0)@W2 + b)
template <int CI, int CO>
__global__ void cheb_combine_kernel(const float* __restrict__ tx0,
                                    const float* __restrict__ tx1,
                                    const float* __restrict__ t2tmp,
                                    const float* __restrict__ W,
                                    const float* __restrict__ b,
                                    float* __restrict__ out, int N) {
  size_t total = (size_t)N * CO;
  for (size_t idx = (size_t)blockIdx.x * blockDim.x + threadIdx.x; idx < total;
       idx += (size_t)gridDim.x * blockDim.x) {
    size_t i = idx / CO;
    int o = (int)(idx % CO);
    float acc = b[o];
#pragma unroll
    for (int ci = 0; ci < CI; ++ci) {
      float a0 = tx0[i * CI + ci];
      float a1 = tx1[i * CI + ci];
      float a2 = 2.0f * t2tmp[i * CI + ci] - a0;
      acc += a0 * W[0 * CI * CO + ci * CO + o];
      acc += a1 * W[1 * CI * CO + ci * CO + o];
      acc += a2 * W[2 * CI * CO + ci * CO + o];
    }
    out[idx] = fmaxf(acc, 0.0f);
  }
}

// ----------------------------------------- cheb layer 3 combine (16->64 WMMA)
// x3 = relu(Tx0@W0 + Tx1@W1 + (2*T2-Tx0)@W2 + b), W: (3,16,64) contiguous
__global__ __launch_bounds__(128) void cheb3_wmma_kernel(
    const float* __restrict__ tx0, const float* __restrict__ tx1,
    const float* __restrict__ t2t, const float* __restrict__ W,
    const float* __restrict__ b, float* __restrict__ out, int N) {
  __shared__ float sW[3][16][64];
#if HAVE_TDM
  if (threadIdx.x < 32) {  // wave 0 drives the TDM DMA; TENSORcnt is per-wave
    tdm_load_1d_to_lds(W, &sW[0][0][0], 3u * 16u * 64u);
    __builtin_amdgcn_s_wait_tensorcnt(0);
  }
  // The TDM write to LDS is invisible to alias analysis (descriptor carries a
  // raw address). Force the compiler to treat sW as written.
  asm volatile("" : : "r"(&sW[0][0][0]) : "memory");
#else
  for (int i = threadIdx.x; i < 3 * 16 * 64; i += 128) ((float*)sW)[i] = W[i];
#endif
  __syncthreads();

  int wave = threadIdx.x >> 5;
  int lane = threadIdx.x & 31;
  int base = (blockIdx.x * 4 + wave) * 16;
  if (base >= N) return;  // uniform per wave

  int m = lane & 15;  // row-in-tile (A) / column (B, D)
  int g = lane >> 4;  // lane-group: K sub-pair select
  int node = base + m;
  if (node > N - 1) node = N - 1;  // clamp loads; EXEC stays full for WMMA
  size_t r0 = (size_t)node * 16;

  v8f acc[4];
#pragma unroll
  for (int nt = 0; nt < 4; ++nt) acc[nt] = vzero8();

#pragma unroll
  for (int k0 = 0; k0 < 16; k0 += 4) {
    int ka = k0 + 2 * g;
    v2f a0, a1, a2;
    a0.x = tx0[r0 + ka];
    a0.y = tx0[r0 + ka + 1];
    a1.x = tx1[r0 + ka];
    a1.y = tx1[r0 + ka + 1];
    a2.x = 2.0f * t2t[r0 + ka] - a0.x;
    a2.y = 2.0f * t2t[r0 + ka + 1] - a0.y;
#pragma unroll
    for (int nt = 0; nt < 4; ++nt) {
      int n = nt * 16 + m;
      v2f b0, b1, b2;
      b0.x = sW[0][ka][n];  b0.y = sW[0][ka + 1][n];
      b1.x = sW[1][ka][n];  b1.y = sW[1][ka + 1][n];
      b2.x = sW[2][ka][n];  b2.y = sW[2][ka + 1][n];
      acc[nt] = wmma_f32_k4(a0, b0, acc[nt]);
      acc[nt] = wmma_f32_k4(a1, b1, acc[nt]);
      acc[nt] = wmma_f32_k4(a2, b2, acc[nt]);
    }
  }

#pragma unroll
  for (int nt = 0; nt < 4; ++nt) {
#pragma unroll
    for (int r = 0; r < 8; ++r) {
      int mm = r + g * 8;
      int nd = base + mm;
      int nn = nt * 16 + m;
      if (nd < N) out[(size_t)nd * 64 + nn] = fmaxf(acc[nt][r] + b[nn], 0.0f);
    }
  }
}

// ----------------------------------------------------------- concat (N x 88)
// xs = [x1(4) | x2(16) | x3(64) | x4(1) | pad(3)]
__global__ void concat_kernel(const float* __restrict__ x1,
                              const float* __restrict__ x2,
                              const float* __restrict__ x3,
                              const float* __restrict__ x4,
                              float* __restrict__ xs, int N) {
  size_t total = (size_t)N * 88;
  for (size_t idx = (size_t)blockIdx.x * blockDim.x + threadIdx.x; idx < total;
       idx += (size_t)gridDim.x * blockDim.x) {
    size_t i = idx / 88;
    int f = (int)(idx % 88);
    float v;
    if (f < 4)        v = x1[i * 4 + f];
    else if (f < 20)  v = x2[i * 16 + (f - 4)];
    else if (f < 84)  v = x3[i * 64 + (f - 20)];
    else if (f == 84) v = x4[i];
    else              v = 0.0f;
    xs[idx] = v;
  }
}

// ------------------------------------------------------- lin1: 85->40 (WMMA)
// LDS layout is zero-padded to 88x48 (not expressible with TDM pow2 pad
// intervals) -> cooperative staging here.
__global__ __launch_bounds__(128) void lin1_wmma_kernel(
    const float* __restrict__ xs /*N*88*/, const float* __restrict__ W /*85*40*/,
    const float* __restrict__ b, float* __restrict__ h1 /*N*40*/, int N) {
  __shared__ float sW[88][48];
  for (int i = threadIdx.x; i < 88 * 48; i += 128) {
    int k = i / 48, n = i % 48;
    sW[k][n] = (k < 85 && n < 40) ? W[k * 40 + n] : 0.0f;
  }
  __syncthreads();

  int wave = threadIdx.x >> 5;
  int lane = threadIdx.x & 31;
  int base = (blockIdx.x * 4 + wave) * 16;
  if (base >= N) return;

  int m = lane & 15;
  int g = lane >> 4;
  int node = base + m;
  if (node > N - 1) node = N - 1;
  size_t r0 = (size_t)node * 88;

  v8f acc[3];
#pragma unroll
  for (int nt = 0; nt < 3; ++nt) acc[nt] = vzero8();

#pragma unroll
  for (int k0 = 0; k0 < 88; k0 += 4) {
    int ka = k0 + 2 * g;
    v2f a;
    a.x = xs[r0 + ka];
    a.y = xs[r0 + ka + 1];
#pragma unroll
    for (int nt = 0; nt < 3; ++nt) {
      int n = nt * 16 + m;
      v2f bb;
      bb.x = sW[ka][n];
      bb.y = sW[ka + 1][n];
      acc[nt] = wmma_f32_k4(a, bb, acc[nt]);
    }
  }

#pragma unroll
  for (int nt = 0; nt < 3; ++nt) {
#pragma unroll
    for (int r = 0; r < 8; ++r) {
      int mm = r + g * 8;
      int nd = base + mm;
      int nn = nt * 16 + m;
      if (nd < N && nn < 40)
        h1[(size_t)nd * 40 + nn] = fmaxf(acc[nt][r] + b[nn], 0.0f);
    }
  }
}

// ------------------------------------------------------- lin2: 40->16 (WMMA)
__global__ __launch_bounds__(128) void lin2_wmma_kernel(
    const float* __restrict__ h1 /*N*40*/, const float* __restrict__ W /*40*16*/,
    const float* __restrict__ b, float* __restrict__ h2 /*N*16*/, int N) {
  __shared__ float sW[40][16];
#if HAVE_TDM
  if (threadIdx.x < 32) {
    tdm_load_1d_to_lds(W, &sW[0][0], 40u * 16u);
    __builtin_amdgcn_s_wait_tensorcnt(0);
  }
  asm volatile("" : : "r"(&sW[0][0]) : "memory");
#else
  for (int i = threadIdx.x; i < 40 * 16; i += 128) ((float*)sW)[i] = W[i];
#endif
  __syncthreads();

  int wave = threadIdx.x >> 5;
  int lane = threadIdx.x & 31;
  int base = (blockIdx.x * 4 + wave) * 16;
  if (base >= N) return;

  int m = lane & 15;
  int g = lane >> 4;
  int node = base + m;
  if (node > N - 1) node = N - 1;
  size_t r0 = (size_t)node * 40;

  v8f acc = vzero8();
#pragma unroll
  for (int k0 = 0; k0 < 40; k0 += 4) {
    int ka = k0 + 2 * g;
    v2f a, bb;
    a.x = h1[r0 + ka];
    a.y = h1[r0 + ka + 1];
    bb.x = sW[ka][m];
    bb.y = sW[ka + 1][m];
    acc = wmma_f32_k4(a, bb, acc);
  }

#pragma unroll
  for (int r = 0; r < 8; ++r) {
    int mm = r + g * 8;
    int nd = base + mm;
    if (nd < N) h2[(size_t)nd * 16 + m] = fmaxf(acc[r] + b[m], 0.0f);
  }
}

// ----------------------------------------------------------- lin3: 16->1
__global__ void lin3_kernel(const float* __restrict__ h2,
                            const float* __restrict__ W,
                            const float* __restrict__ b,
                            float* __restrict__ xo, int N) {
  for (size_t i = (size_t)blockIdx.x * blockDim.x + threadIdx.x; i < (size_t)N;
       i += (size_t)gridDim.x * blockDim.x) {
    float acc = b[0];
#pragma unroll
    for (int c = 0; c < 16; ++c) acc += h2[i * 16 + c] * W[c];
    xo[i] = fmaxf(acc, 0.0f);
  }
}

// ----------------------------------------------------------- edge MLP (x2)
static __device__ __forceinline__ float edge_stage(
    float s, float d, float a, const float* __restrict__ w1,
    const float* __restrict__ b1, const float* __restrict__ g,
    const float* __restrict__ be, const float* __restrict__ w2,
    const float* __restrict__ b2) {
  float h0 = s * w1[0] + d * w1[2] + a * w1[4] + b1[0];
  float h1 = s * w1[1] + d * w1[3] + a * w1[5] + b1[1];
  h0 = fmaxf(h0, 0.0f);
  h1 = fmaxf(h1, 0.0f);
  float m = 0.5f * (h0 + h1);
  float d0 = h0 - m, d1 = h1 - m;
  float var = 0.5f * (d0 * d0 + d1 * d1);
  float r = rsqrtf(var + EPS);
  float n0 = d0 * r * g[0] + be[0];
  float n1 = d1 * r * g[1] + be[1];
  float o = n0 * w2[0] + n1 * w2[1] + b2[0];
  return fmaxf(o, 0.0f);  // outer relu in reference
}

__global__ void edge_mlp_kernel(
    const int* __restrict__ row, const int* __restrict__ col,
    const float* __restrict__ eattr, const float* __restrict__ xo,
    const float* __restrict__ w1a, const float* __restrict__ b1a,
    const float* __restrict__ ga, const float* __restrict__ bea,
    const float* __restrict__ w2a, const float* __restrict__ b2a,
    const float* __restrict__ w1b, const float* __restrict__ b1b,
    const float* __restrict__ gb, const float* __restrict__ beb,
    const float* __restrict__ w2b, const float* __restrict__ b2b,
    float* __restrict__ out, int E) {
  for (size_t e = (size_t)blockIdx.x * blockDim.x + threadIdx.x; e < (size_t)E;
       e += (size_t)gridDim.x * blockDim.x) {
    float s = xo[row[e]];
    float d = xo[col[e]];
    float a = eattr[e];
    float v = edge_stage(s, d, a, w1a, b1a, ga, bea, w2a, b2a);
    v = edge_stage(s, d, v, w1b, b1b, gb, beb, w2b, b2b);
    out[e] = v;
  }
}

// ---------------------------------------------------------------- launcher
static inline unsigned nblocks(size_t total) {
  size_t b = (total + 255) / 256;
  if (b > (size_t)1 << 20) b = (size_t)1 << 20;
  if (b == 0) b = 1;
  return (unsigned)b;
}

extern "C" void kernel_launch(void* const* d_in, const int* in_sizes, int n_in,
                              void* d_out, int out_size, void* d_ws,
                              size_t ws_size, hipStream_t stream) {
  (void)n_in; (void)out_size; (void)ws_size;

  const float* x     = (const float*)d_in[0];
  const int*   ei    = (const int*)d_in[1];
  const float* eattr = (const float*)d_in[2];
  const float* c1W = (const float*)d_in[3];  const float* c1b = (const float*)d_in[4];
  const float* c2W = (const float*)d_in[5];  const float* c2b = (const float*)d_in[6];
  const float* c3W = (const float*)d_in[7];  const float* c3b = (const float*)d_in[8];
  const float* c4W = (const float*)d_in[9];  const float* c4b = (const float*)d_in[10];
  const float* l1W = (const float*)d_in[11]; const float* l1b = (const float*)d_in[12];
  const float* l2W = (const float*)d_in[13]; const float* l2b = (const float*)d_in[14];
  const float* l3W = (const float*)d_in[15]; const float* l3b = (const float*)d_in[16];
  const float* e1w1 = (const float*)d_in[17]; const float* e1b1 = (const float*)d_in[18];
  const float* e1g  = (const float*)d_in[19]; const float* e1be = (const float*)d_in[20];
  const float* e1w2 = (const float*)d_in[21]; const float* e1b2 = (const float*)d_in[22];
  const float* e2w1 = (const float*)d_in[23]; const float* e2b1 = (const float*)d_in[24];
  const float* e2g  = (const float*)d_in[25]; const float* e2be = (const float*)d_in[26];
  const float* e2w2 = (const float*)d_in[27]; const float* e2b2 = (const float*)d_in[28];

  const int N = in_sizes[0];
  const int E = in_sizes[2];
  const int* row = ei;
  const int* col = ei + E;

  // workspace layout (floats)
  float* ws = (float*)d_ws;
  size_t off = 0;
  float* deg   = ws + off; off += (size_t)N;
  float* wn    = ws + off; off += (size_t)E;
  float* xn    = ws + off; off += (size_t)N * 64;
  float* t1    = ws + off; off += (size_t)N * 64;
  float* t2    = ws + off; off += (size_t)N * 64;
  float* x1    = ws + off; off += (size_t)N * 4;
  float* x2    = ws + off; off += (size_t)N * 16;
  float* x3    = ws + off; off += (size_t)N * 64;
  float* x4    = ws + off; off += (size_t)N;
  float* xs    = ws + off; off += (size_t)N * 88;
  float* h1    = ws + off; off += (size_t)N * 40;
  float* h2    = ws + off; off += (size_t)N * 16;
  float* stats = ws + off; off += 128;

  float* xo = (float*)d_out;  // N
  float* ea = xo + N;         // E

  const unsigned wtiles = (unsigned)(((N + 15) / 16 + 3) / 4);  // 4 waves/block

  // graph normalization weights
  fillzero_kernel<<<nblocks(N), 256, 0, stream>>>(deg, (size_t)N);
  deg_kernel<<<nblocks(E), 256, 0, stream>>>(row, eattr, deg, E);
  wn_kernel<<<nblocks(E), 256, 0, stream>>>(row, col, eattr, deg, wn, E);

  // ---- cheb layer 1: 1 -> 4
  fillzero_kernel<<<1, 256, 0, stream>>>(stats, 128);
  inorm_stats_kernel<1><<<nblocks(N), 256, 0, stream>>>(x, stats, N);
  inorm_apply_kernel<1><<<nblocks(N), 256, 0, stream>>>(x, stats, xn, N);
  fillzero_kernel<<<nblocks(N), 256, 0, stream>>>(t1, (size_t)N);
  spmm_kernel<1><<<nblocks(E), 256, 0, stream>>>(row, col, wn, xn, t1, E);
  fillzero_kernel<<<nblocks(N), 256, 0, stream>>>(t2, (size_t)N);
  spmm_kernel<1><<<nblocks(E), 256, 0, stream>>>(row, col, wn, t1, t2, E);
  cheb_combine_kernel<1, 4><<<nblocks((size_t)N * 4), 256, 0, stream>>>(
      xn, t1, t2, c1W, c1b, x1, N);

  // ---- cheb layer 2: 4 -> 16
  fillzero_kernel<<<1, 256, 0, stream>>>(stats, 128);
  inorm_stats_kernel<4><<<nblocks((size_t)N * 4), 256, 0, stream>>>(x1, stats, N);
  inorm_apply_kernel<4><<<nblocks((size_t)N * 4), 256, 0, stream>>>(x1, stats, xn, N);
  fillzero_kernel<<<nblocks((size_t)N * 4), 256, 0, stream>>>(t1, (size_t)N * 4);
  spmm_kernel<4><<<nblocks((size_t)E * 4), 256, 0, stream>>>(row, col, wn, xn, t1, E);
  fillzero_kernel<<<nblocks((size_t)N * 4), 256, 0, stream>>>(t2, (size_t)N * 4);
  spmm_kernel<4><<<nblocks((size_t)E * 4), 256, 0, stream>>>(row, col, wn, t1, t2, E);
  cheb_combine_kernel<4, 16><<<nblocks((size_t)N * 16), 256, 0, stream>>>(
      xn, t1, t2, c2W, c2b, x2, N);

  // ---- cheb layer 3: 16 -> 64 (WMMA combine, TDM weight staging)
  fillzero_kernel<<<1, 256, 0, stream>>>(stats, 128);
  inorm_stats_kernel<16><<<nblocks((size_t)N * 16), 256, 0, stream>>>(x2, stats, N);
  inorm_apply_kernel<16><<<nblocks((size_t)N * 16), 256, 0, stream>>>(x2, stats, xn, N);
  fillzero_kernel<<<nblocks((size_t)N * 16), 256, 0, stream>>>(t1, (size_t)N * 16);
  spmm_kernel<16><<<nblocks((size_t)E * 16), 256, 0, stream>>>(row, col, wn, xn, t1, E);
  fillzero_kernel<<<nblocks((size_t)N * 16), 256, 0, stream>>>(t2, (size_t)N * 16);
  spmm_kernel<16><<<nblocks((size_t)E * 16), 256, 0, stream>>>(row, col, wn, t1, t2, E);
  cheb3_wmma_kernel<<<wtiles, 128, 0, stream>>>(xn, t1, t2, c3W, c3b, x3, N);

  // ---- cheb layer 4: 64 -> 1
  fillzero_kernel<<<1, 256, 0, stream>>>(stats, 128);
  inorm_stats_kernel<64><<<nblocks((size_t)N * 64), 256, 0, stream>>>(x3, stats, N);
  inorm_apply_kernel<64><<<nblocks((size_t)N * 64), 256, 0, stream>>>(x3, stats, xn, N);
  fillzero_kernel<<<nblocks((size_t)N * 64), 256, 0, stream>>>(t1, (size_t)N * 64);
  spmm_kernel<64><<<nblocks((size_t)E * 64), 256, 0, stream>>>(row, col, wn, xn, t1, E);
  fillzero_kernel<<<nblocks((size_t)N * 64), 256, 0, stream>>>(t2, (size_t)N * 64);
  spmm_kernel<64><<<nblocks((size_t)E * 64), 256, 0, stream>>>(row, col, wn, t1, t2, E);
  cheb_combine_kernel<64, 1><<<nblocks(N), 256, 0, stream>>>(
      xn, t1, t2, c4W, c4b, x4, N);

  // ---- node MLP head (WMMA)
  concat_kernel<<<nblocks((size_t)N * 88), 256, 0, stream>>>(x1, x2, x3, x4, xs, N);
  lin1_wmma_kernel<<<wtiles, 128, 0, stream>>>(xs, l1W, l1b, h1, N);
  lin2_wmma_kernel<<<wtiles, 128, 0, stream>>>(h1, l2W, l2b, h2, N);
  lin3_kernel<<<nblocks(N), 256, 0, stream>>>(h2, l3W, l3b, xo, N);

  // ---- edge MLPs (fused, per-edge registers)
  edge_mlp_kernel<<<nblocks(E), 256, 0, stream>>>(
      row, col, eattr, xo, e1w1, e1b1, e1g, e1be, e1w2, e1b2, e2w1, e2b1, e2g,
      e2be, e2w2, e2b2, ea, E);
}